// lightGRAND_NX_20693152432219
// MI455X (gfx1250) — compile-verified
//
#include <hip/hip_runtime.h>
#include <math.h>

typedef __attribute__((ext_vector_type(2))) float v2f;
typedef __attribute__((ext_vector_type(8))) float v8f;

#define IN_DIM 128
#define HID    256
#define NCLASS 40

// ---------------- x = feats * 0.5 -> h0 and y ----------------
__global__ __launch_bounds__(256) void k_init(const float* __restrict__ feats,
                                              float* __restrict__ h0,
                                              float* __restrict__ y, int n4) {
  int i = blockIdx.x * 256 + threadIdx.x;
  if (i >= n4) return;
  float4 f = ((const float4*)feats)[i];
  float4 o;
  o.x = f.x * 0.5f; o.y = f.y * 0.5f; o.z = f.z * 0.5f; o.w = f.w * 0.5f;
  ((float4*)h0)[i] = o;
  ((float4*)y)[i]  = o;
}

__global__ __launch_bounds__(256) void k_zero(float* __restrict__ p, int n4) {
  int i = blockIdx.x * 256 + threadIdx.x;
  if (i >= n4) return;
  float4 z; z.x = z.y = z.z = z.w = 0.f;
  ((float4*)p)[i] = z;
}

__global__ __launch_bounds__(256) void k_axpy(float* __restrict__ y,
                                              const float* __restrict__ h, int n4) {
  int i = blockIdx.x * 256 + threadIdx.x;
  if (i >= n4) return;
  float4 a = ((const float4*)y)[i];
  float4 b = ((const float4*)h)[i];
  a.x += b.x; a.y += b.y; a.z += b.z; a.w += b.w;
  ((float4*)y)[i] = a;
}

// ---------------- SpMM: out[row] += val * x[col]  (one wave32 per edge) ----------------
__global__ __launch_bounds__(256) void k_spmm(const int* __restrict__ rows,
                                              const int* __restrict__ cols,
                                              const float* __restrict__ vals,
                                              const float* __restrict__ x,
                                              float* __restrict__ out, int E) {
  int e = blockIdx.x * 8 + (threadIdx.x >> 5);
  if (e >= E) return;
  int lane = threadIdx.x & 31;
  int r = rows[e];
  int c = cols[e];
  float v = vals[e];
  float4 p = ((const float4*)(x + (size_t)c * IN_DIM))[lane]; // 32 lanes * 4 = 128 dims
  float* o = out + (size_t)r * IN_DIM + lane * 4;
  atomicAdd(o + 0, v * p.x);
  atomicAdd(o + 1, v * p.y);
  atomicAdd(o + 2, v * p.z);
  atomicAdd(o + 3, v * p.w);
}

// ---------------- GEMM1: h1 = relu((y * 0.25) @ W1 + b1) via fp32 WMMA ----------------
// grid (2, N/16), block 256 (8 waves). Wave w computes 16x16 tile at
// rows [mtile*16, +16), cols [(blockIdx.x*8+w)*16, +16).
__global__ __launch_bounds__(256) void k_gemm1(const float* __restrict__ y,
                                               const float* __restrict__ W1,
                                               const float* __restrict__ b1,
                                               float* __restrict__ h1) {
  __shared__ float As[16 * 132]; // stride 132 to dodge 64-bank conflicts on stride-128
  const int mtile = blockIdx.y;
  const int tid = threadIdx.x;
  // cooperative load of 16x128 A tile (pre-scaled by 1/(ORDER+1) = 0.25)
  for (int i = tid; i < 512; i += 256) {            // 512 float4 = 2048 floats
    int row = i >> 5;
    int c4  = (i & 31) << 2;
    float4 f = *(const float4*)(y + ((size_t)(mtile * 16 + row)) * IN_DIM + c4);
    float* d = &As[row * 132 + c4];
    d[0] = f.x * 0.25f; d[1] = f.y * 0.25f; d[2] = f.z * 0.25f; d[3] = f.w * 0.25f;
  }
  __syncthreads();

  const int wave = tid >> 5;
  const int lane = tid & 31;
  const int n0 = (blockIdx.x * 8 + wave) * 16;
  const int mn = lane & 15;          // A: row M; B/C: col N
  const int kb = (lane >> 4) * 2;    // K sub-offset per half-wave
  const int n  = n0 + mn;

  v8f c = {};
  for (int k = 0; k < IN_DIM; k += 4) {
    v2f a, b;
    a.x = As[mn * 132 + k + kb];
    a.y = As[mn * 132 + k + kb + 1];
    b.x = W1[(k + kb) * HID + n];
    b.y = W1[(k + kb + 1) * HID + n];
    c = __builtin_amdgcn_wmma_f32_16x16x4_f32(false, a, false, b, (short)0, c,
                                              false, false);
  }
  const float bias = b1[n];
  const int half = (lane >> 4) * 8;  // C layout: VGPR r -> M = r + 8*(lane>=16)
#pragma unroll
  for (int r = 0; r < 8; ++r) {
    int row = mtile * 16 + r + half;
    h1[(size_t)row * HID + n] = fmaxf(c[r] + bias, 0.f);
  }
}

// ---------------- GEMM2: logits = h1 @ W2 + b2 via fp32 WMMA ----------------
// grid (N/16), block 96 (3 waves covering 48 padded cols, guarded at 40).
__global__ __launch_bounds__(96) void k_gemm2(const float* __restrict__ h1,
                                              const float* __restrict__ W2,
                                              const float* __restrict__ b2,
                                              float* __restrict__ logits) {
  const int mtile = blockIdx.x;
  const int wave = threadIdx.x >> 5;
  const int lane = threadIdx.x & 31;
  const int n0 = wave * 16;
  const int mn = lane & 15;
  const int kb = (lane >> 4) * 2;
  const int col = n0 + mn;

  const float* arow = h1 + (size_t)(mtile * 16 + mn) * HID;
  v8f c = {};
  for (int k = 0; k < HID; k += 4) {
    v2f a = *(const v2f*)(arow + k + kb);   // 8B-aligned contiguous pair
    v2f b;
    b.x = (col < NCLASS) ? W2[(k + kb) * NCLASS + col] : 0.f;
    b.y = (col < NCLASS) ? W2[(k + kb + 1) * NCLASS + col] : 0.f;
    c = __builtin_amdgcn_wmma_f32_16x16x4_f32(false, a, false, b, (short)0, c,
                                              false, false);
  }
  if (col < NCLASS) {
    const float bias = b2[col];
    const int half = (lane >> 4) * 8;
#pragma unroll
    for (int r = 0; r < 8; ++r) {
      int row = mtile * 16 + r + half;
      logits[(size_t)row * NCLASS + col] = c[r] + bias;
    }
  }
}

// ---------------- double log_softmax over 40 classes (one wave32 per row) ----------------
__global__ __launch_bounds__(256) void k_lsm2(const float* __restrict__ logits,
                                              float* __restrict__ out, int N) {
  int row = blockIdx.x * 8 + (threadIdx.x >> 5);
  if (row >= N) return;
  int lane = threadIdx.x & 31;
  const float* lp = logits + (size_t)row * NCLASS;
  float v0 = lp[lane];                                       // lanes 0..31
  float v1 = (lane < 8) ? lp[32 + lane] : -__builtin_inff(); // lanes carry cls 32..39
#pragma unroll
  for (int pass = 0; pass < 2; ++pass) {
    float mx = fmaxf(v0, v1);
    for (int off = 16; off > 0; off >>= 1)
      mx = fmaxf(mx, __shfl_xor(mx, off, 32));
    float s = expf(v0 - mx) + ((lane < 8) ? expf(v1 - mx) : 0.f);
    for (int off = 16; off > 0; off >>= 1)
      s += __shfl_xor(s, off, 32);
    float lse = logf(s);
    v0 = v0 - mx - lse;
    v1 = v1 - mx - lse;
  }
  out[(size_t)row * NCLASS + lane] = v0;
  if (lane < 8) out[(size_t)row * NCLASS + 32 + lane] = v1;
}

extern "C" void kernel_launch(void* const* d_in, const int* in_sizes, int n_in,
                              void* d_out, int out_size, void* d_ws, size_t ws_size,
                              hipStream_t stream) {
  const float* feats = (const float*)d_in[0];
  const int*   erow  = (const int*)d_in[1];
  const int*   ecol  = (const int*)d_in[2];
  const float* evals = (const float*)d_in[3];
  const float* W1    = (const float*)d_in[4];
  const float* b1    = (const float*)d_in[5];
  const float* W2    = (const float*)d_in[6];
  const float* b2    = (const float*)d_in[7];
  float* out = (float*)d_out;

  const int N = in_sizes[0] / IN_DIM;   // 50000 (= 3125 * 16)
  const int E = in_sizes[1];            // 800000
  const int nfeat = N * IN_DIM;         // 6.4M floats
  const int n4 = nfeat / 4;

  // workspace layout (needs 3 * N*128*4 = 76.8 MB):
  //   buf0,buf1: ping-pong hop buffers; after hops, h1 (N*256) overlays buf0+buf1
  //   buf2: running sum y; after GEMM1, logits (N*40) overlay buf2
  float* buf0 = (float*)d_ws;
  float* buf1 = buf0 + nfeat;
  float* buf2 = buf1 + nfeat;
  float* h1     = buf0;
  float* logits = buf2;

  dim3 b256(256);
  int g4 = (n4 + 255) / 256;

  k_init<<<g4, b256, 0, stream>>>(feats, buf0, buf2, n4);

  float* cur = buf0;
  float* nxt = buf1;
  for (int hop = 0; hop < 3; ++hop) {
    k_zero<<<g4, b256, 0, stream>>>(nxt, n4);
    k_spmm<<<(E + 7) / 8, b256, 0, stream>>>(erow, ecol, evals, cur, nxt, E);
    k_axpy<<<g4, b256, 0, stream>>>(buf2, nxt, n4);
    float* t = cur; cur = nxt; nxt = t;
  }

  const int mtiles = N / 16;
  k_gemm1<<<dim3(2, mtiles), b256, 0, stream>>>(buf2, W1, b1, h1);
  k_gemm2<<<dim3(mtiles), dim3(96), 0, stream>>>(h1, W2, b2, logits);
  k_lsm2<<<(N + 7) / 8, b256, 0, stream>>>(logits, out, N);
}